// SPCENTrain_19009525252258
// MI455X (gfx1250) — compile-verified
//
#include <hip/hip_runtime.h>
#include <hip/hip_bf16.h>

// PCEN via chunked parallel linear scan + gfx1250 async global->LDS streaming.
//
//   m_t = (1-s) m_{t-1} + s x_t          (linear recurrence, per mel)
//   y_t = (x_t * (m_t+eps)^-alpha + delta)^r - delta^r
//
// Pass 1: per-chunk zero-init EMA end state S[b][c][m]      (reads x)
// Pass 2: exclusive scan across chunks: P[c] = a^L P[c-1] + S[c-1]
// Pass 3: replay each chunk seeded with P, emit outputs      (reads x, writes y)

namespace {

constexpr int B_ = 64;
constexpr int T_ = 8000;
constexpr int M_ = 80;
constexpr int C_ = 50;            // time chunks
constexpr int L_ = T_ / C_;       // 160 timesteps / chunk
constexpr int TILE = 32;          // timesteps per LDS tile
constexpr int NTILES = L_ / TILE; // 5 tiles / chunk
constexpr int BLK = 128;          // 4 waves
constexpr int KCH = (TILE * M_ * 4) / 16 / BLK; // 5 x b128 per lane per tile
constexpr float EPS_ = 1e-6f;

static_assert(KCH * BLK * 16 == TILE * M_ * 4, "tile copy must be exact");

__device__ __forceinline__ void async_b128(unsigned lds_byte, const float* gsrc) {
  // gfx1250: VDST = LDS byte address, VADDR = 64-bit global address.
  asm volatile("global_load_async_to_lds_b128 %0, %1, off"
               :: "v"(lds_byte), "v"(gsrc)
               : "memory");
}

template <int N>
__device__ __forceinline__ void wait_asynccnt() {
#if __has_builtin(__builtin_amdgcn_s_wait_asynccnt)
  __builtin_amdgcn_s_wait_asynccnt(N);
#else
  asm volatile("s_wait_asynccnt %0" :: "n"(N) : "memory");
#endif
}

template <bool WRITE_OUT>
__global__ __launch_bounds__(BLK)
void pcen_scan_kernel(const float* __restrict__ x,
                      const float* __restrict__ log_s,
                      const float* __restrict__ log_alpha,
                      const float* __restrict__ log_delta,
                      const float* __restrict__ log_r,
                      const float* __restrict__ Pin,   // chunk seeds (pass 3)
                      float* __restrict__ outS) {      // S (pass 1) or y (pass 3)
  __shared__ float smem[2][TILE * M_];

  const int tid = threadIdx.x;
  const int b = blockIdx.x / C_;
  const int c = blockIdx.x % C_;
  const float* __restrict__ xg = x + ((size_t)b * T_ + (size_t)c * L_) * M_;

  // Per-mel parameters for compute lanes.
  float s = 0.f, aa = 0.f, alpha = 0.f, delta = 0.f, r = 0.f, dr = 0.f, mst = 0.f;
  if (tid < M_) {
    s  = __expf(log_s[tid]);
    aa = 1.0f - s;
    if (WRITE_OUT) {
      alpha = __expf(log_alpha[tid]);
      delta = __expf(log_delta[tid]);
      r     = __expf(log_r[tid]);
      dr    = __expf(r * __logf(delta));             // delta^r
      mst   = Pin[((size_t)b * C_ + c) * M_ + tid];  // incoming chunk state
    }
  }

  // Issue one tile's async copy: BLK lanes x KCH b128s = TILE*M_ floats.
  auto issue_tile = [&](int tile, int buf) {
    const float* gsrc = xg + (size_t)tile * TILE * M_;
    const unsigned lbase = (unsigned)(size_t)&smem[buf][0];
#pragma unroll
    for (int k = 0; k < KCH; ++k) {
      const int j = tid + k * BLK;                  // 16-byte chunk index
      async_b128(lbase + (unsigned)j * 16u, gsrc + (size_t)j * 4);
    }
  };

  issue_tile(0, 0);  // prologue

  for (int i = 0; i < NTILES; ++i) {
    if (i + 1 < NTILES) {
      issue_tile(i + 1, (i + 1) & 1);
      wait_asynccnt<KCH>();   // in-order completion => tile i landed
    } else {
      wait_asynccnt<0>();
    }
    __syncthreads();          // all waves' portions visible

    if (tid < M_) {
      const float* __restrict__ bufp = &smem[i & 1][0];
      float* __restrict__ op =
          WRITE_OUT ? (outS + ((size_t)b * T_ + (size_t)(c * L_ + i * TILE)) * M_ + tid)
                    : nullptr;
#pragma unroll
      for (int t = 0; t < TILE; ++t) {
        const float xv = bufp[t * M_ + tid];
        mst = fmaf(aa, mst, s * xv);                // EMA step
        if (WRITE_OUT) {
          const float mc   = mst + EPS_;
          const float inv  = __expf(-alpha * __logf(mc));  // (m+eps)^-alpha
          const float base = fmaf(xv, inv, delta);
          op[(size_t)t * M_] = __expf(r * __logf(base)) - dr;
        }
      }
    }
    __syncthreads();          // buffer safe for reuse by next async copy
  }

  if (!WRITE_OUT && tid < M_) {
    outS[((size_t)b * C_ + c) * M_ + tid] = mst;    // chunk-local end state
  }
}

__global__ void pcen_prefix_kernel(const float* __restrict__ log_s,
                                   const float* __restrict__ S,
                                   float* __restrict__ P) {
  const int id = blockIdx.x * blockDim.x + threadIdx.x;
  if (id >= B_ * M_) return;
  const int b = id / M_;
  const int m = id % M_;
  const float s  = __expf(log_s[m]);
  const float a  = 1.0f - s;
  const float aL = __expf((float)L_ * __logf(a));   // a^L
  float p = 0.f;
  for (int c = 0; c < C_; ++c) {
    const size_t idx = ((size_t)b * C_ + c) * M_ + m;
    P[idx] = p;                                     // state at chunk start
    p = fmaf(aL, p, S[idx]);                        // carry to next chunk
  }
}

} // namespace

extern "C" void kernel_launch(void* const* d_in, const int* in_sizes, int n_in,
                              void* d_out, int out_size, void* d_ws, size_t ws_size,
                              hipStream_t stream) {
  (void)in_sizes; (void)n_in; (void)out_size; (void)ws_size;

  const float* x         = (const float*)d_in[0];
  const float* log_s     = (const float*)d_in[1];
  const float* log_alpha = (const float*)d_in[2];
  const float* log_delta = (const float*)d_in[3];
  const float* log_r     = (const float*)d_in[4];
  float* out = (float*)d_out;

  float* S = (float*)d_ws;                          // [B][C][M] chunk end states
  float* P = S + (size_t)B_ * C_ * M_;              // [B][C][M] chunk seeds

  // Pass 1: per-chunk local EMA end states (streams x via async LDS copies).
  pcen_scan_kernel<false><<<B_ * C_, BLK, 0, stream>>>(
      x, log_s, log_alpha, log_delta, log_r, nullptr, S);

  // Pass 2: tiny cross-chunk exclusive scan.
  pcen_prefix_kernel<<<(B_ * M_ + 255) / 256, 256, 0, stream>>>(log_s, S, P);

  // Pass 3: replay chunks seeded with P, emit PCEN outputs.
  pcen_scan_kernel<true><<<B_ * C_, BLK, 0, stream>>>(
      x, log_s, log_alpha, log_delta, log_r, P, out);
}